// GATLayer_30116310680382
// MI455X (gfx1250) — compile-verified
//
#include <hip/hip_runtime.h>
#include <hip/hip_bf16.h>

#define DIN   128
#define DOUT  64
#define DE    50
#define NEG_SLOPE 0.01f

typedef __attribute__((ext_vector_type(2))) float v2f;
typedef __attribute__((ext_vector_type(8))) float v8f;

// ---------------------------------------------------------------------------
// init: nh = 0, m = -inf, denom = 0
// ---------------------------------------------------------------------------
__global__ void gat_init(float* __restrict__ nh, float* __restrict__ m,
                         float* __restrict__ denom, int N) {
    int i = blockIdx.x * blockDim.x + threadIdx.x;
    if (i < N * DOUT) nh[i] = 0.0f;
    if (i < N) { m[i] = -__builtin_inff(); denom[i] = 0.0f; }
}

// ---------------------------------------------------------------------------
// z = h @ W^T via V_WMMA_F32_16X16X4_F32.
// One wave computes a 16-row x 64-col tile of z (4 accumulators of 16x16).
// f32 A layout (ISA 7.12.2): lane = lm + 16*hi;
//   a.x = A[M=lm, K=k0+2*hi], a.y = A[M=lm, K=k0+2*hi+1]
// f32 B layout mirrors A; since B = W^T, B[k, o] = W[o, k], so per lane:
//   b.x = W[(o0+lm)*DIN + k0+2*hi], b.y = same +1
// C/D layout: lane lm+16*hi, vgpr i  -> D[M = 8*hi + i, N = lm]
// ---------------------------------------------------------------------------
__global__ void __launch_bounds__(256)
gat_gemm_z(const float* __restrict__ h, const float* __restrict__ W,
           float* __restrict__ z, int N) {
    const int lane = threadIdx.x & 31;
    const int wave = threadIdx.x >> 5;
    const int row0 = (blockIdx.x * 8 + wave) * 16;
    if (row0 >= N) return;

    const int lm = lane & 15;
    const int hi = lane >> 4;

    int arow = row0 + lm; if (arow >= N) arow = N - 1;     // tail clamp (loads only)
    const float* __restrict__ ha = h + (size_t)arow * DIN + 2 * hi;
    const float* __restrict__ w0 = W + (size_t)(0 * 16 + lm) * DIN + 2 * hi;
    const float* __restrict__ w1 = W + (size_t)(1 * 16 + lm) * DIN + 2 * hi;
    const float* __restrict__ w2 = W + (size_t)(2 * 16 + lm) * DIN + 2 * hi;
    const float* __restrict__ w3 = W + (size_t)(3 * 16 + lm) * DIN + 2 * hi;

    v8f acc0 = {}, acc1 = {}, acc2 = {}, acc3 = {};

#pragma unroll 4
    for (int k0 = 0; k0 < DIN; k0 += 4) {
        v2f a, b0, b1, b2, b3;
        a.x  = ha[k0]; a.y  = ha[k0 + 1];
        b0.x = w0[k0]; b0.y = w0[k0 + 1];
        b1.x = w1[k0]; b1.y = w1[k0 + 1];
        b2.x = w2[k0]; b2.y = w2[k0 + 1];
        b3.x = w3[k0]; b3.y = w3[k0 + 1];
        acc0 = __builtin_amdgcn_wmma_f32_16x16x4_f32(false, a, false, b0, (short)0, acc0, false, false);
        acc1 = __builtin_amdgcn_wmma_f32_16x16x4_f32(false, a, false, b1, (short)0, acc1, false, false);
        acc2 = __builtin_amdgcn_wmma_f32_16x16x4_f32(false, a, false, b2, (short)0, acc2, false, false);
        acc3 = __builtin_amdgcn_wmma_f32_16x16x4_f32(false, a, false, b3, (short)0, acc3, false, false);
    }

#pragma unroll
    for (int i = 0; i < 8; ++i) {
        int mrow = row0 + hi * 8 + i;
        if (mrow < N) {
            float* __restrict__ zr = z + (size_t)mrow * DOUT;
            zr[0 * 16 + lm] = acc0[i];
            zr[1 * 16 + lm] = acc1[i];
            zr[2 * 16 + lm] = acc2[i];
            zr[3 * 16 + lm] = acc3[i];
        }
    }
}

// ---------------------------------------------------------------------------
// per-node attention scores: s_src[n] = z[n]·w_attn[0:64], s_dst[n] = z[n]·w_attn[114:178]
// one wave32 per node, shuffle reduction
// ---------------------------------------------------------------------------
__global__ void __launch_bounds__(256)
gat_node_scores(const float* __restrict__ z, const float* __restrict__ w_attn,
                float* __restrict__ s_src, float* __restrict__ s_dst, int N) {
    int n    = (blockIdx.x * blockDim.x + threadIdx.x) >> 5;
    int lane = threadIdx.x & 31;
    if (n >= N) return;
    const float* __restrict__ zr = z + (size_t)n * DOUT;
    float z0 = zr[lane], z1 = zr[lane + 32];
    float zs = z0 * w_attn[lane]             + z1 * w_attn[lane + 32];
    float zd = z0 * w_attn[DOUT + DE + lane] + z1 * w_attn[DOUT + DE + lane + 32];
#pragma unroll
    for (int off = 16; off; off >>= 1) {
        zs += __shfl_xor(zs, off, 32);
        zd += __shfl_xor(zd, off, 32);
    }
    if (lane == 0) { s_src[n] = zs; s_dst[n] = zd; }
}

// signed/unsigned-punned float atomic max (valid with -inf init)
__device__ __forceinline__ void atomicMaxF(float* addr, float val) {
    if (!(val < 0.0f)) atomicMax((int*)addr, __float_as_int(val));
    else               atomicMin((unsigned int*)addr, __float_as_uint(val));
}

// ---------------------------------------------------------------------------
// edge pass 1: e = leakyrelu(s_src[src] + y·w_e + s_dst[dst]); segment max into m
// one wave32 per edge (De=50 contiguous floats)
// ---------------------------------------------------------------------------
__global__ void __launch_bounds__(256)
gat_edge1(const float* __restrict__ y, const float* __restrict__ w_attn,
          const float* __restrict__ s_src, const float* __restrict__ s_dst,
          const int* __restrict__ src, const int* __restrict__ dst,
          float* __restrict__ e_buf, float* __restrict__ m, int E) {
    int i    = (blockIdx.x * blockDim.x + threadIdx.x) >> 5;
    int lane = threadIdx.x & 31;
    if (i >= E) return;
    const float* __restrict__ yr = y + (size_t)i * DE;
    float acc = 0.0f;
    if (lane < DE)      acc  = yr[lane]      * w_attn[DOUT + lane];
    if (lane + 32 < DE) acc += yr[lane + 32] * w_attn[DOUT + lane + 32];
#pragma unroll
    for (int off = 16; off; off >>= 1) acc += __shfl_xor(acc, off, 32);
    if (lane == 0) {
        float ev = s_src[src[i]] + acc + s_dst[dst[i]];
        ev = (ev >= 0.0f) ? ev : NEG_SLOPE * ev;
        e_buf[i] = ev;
        atomicMaxF(&m[dst[i]], ev);
    }
}

// ---------------------------------------------------------------------------
// edge pass 2: ex = exp(e - m[dst]); denom[dst] += ex   (in-place e_buf -> ex)
// ---------------------------------------------------------------------------
__global__ void __launch_bounds__(256)
gat_edge2(float* __restrict__ e_buf, const float* __restrict__ m,
          const int* __restrict__ dst, float* __restrict__ denom, int E) {
    int i = blockIdx.x * blockDim.x + threadIdx.x;
    if (i >= E) return;
    int d = dst[i];
    float ex = __expf(e_buf[i] - m[d]);
    e_buf[i] = ex;
    atomicAdd(&denom[d], ex);
}

// ---------------------------------------------------------------------------
// edge pass 3: nh[dst] += (ex / max(denom[dst],1e-38)) * z[src]
// 64 threads per edge (one per output dim); z is L2-resident (26 MB << 192 MB)
// ---------------------------------------------------------------------------
__global__ void __launch_bounds__(256)
gat_edge3(const float* __restrict__ e_buf, const float* __restrict__ denom,
          const float* __restrict__ z, const int* __restrict__ src,
          const int* __restrict__ dst, float* __restrict__ nh, int E) {
    int t = blockIdx.x * blockDim.x + threadIdx.x;
    int i = t >> 6;
    int d = t & 63;
    if (i >= E) return;
    int sn = src[i], dn = dst[i];
    float alpha = e_buf[i] / fmaxf(denom[dn], 1e-38f);
    atomicAdd(&nh[(size_t)dn * DOUT + d], alpha * z[(size_t)sn * DOUT + d]);
}

// ---------------------------------------------------------------------------
extern "C" void kernel_launch(void* const* d_in, const int* in_sizes, int n_in,
                              void* d_out, int out_size, void* d_ws, size_t ws_size,
                              hipStream_t stream) {
    const float* h      = (const float*)d_in[0];
    const float* y      = (const float*)d_in[1];
    const float* W_fc   = (const float*)d_in[2];
    const float* w_attn = (const float*)d_in[3];
    const int*   src    = (const int*)d_in[4];
    const int*   dst    = (const int*)d_in[5];
    float* nh = (float*)d_out;

    const int N = in_sizes[0] / DIN;
    const int E = in_sizes[4];

    // workspace layout (floats): z[N*64] | s_src[N] | s_dst[N] | m[N] | denom[N] | e_buf[E]
    float* z     = (float*)d_ws;
    float* s_src = z + (size_t)N * DOUT;
    float* s_dst = s_src + N;
    float* m     = s_dst + N;
    float* denom = m + N;
    float* e_buf = denom + N;

    int initGrid = (N * DOUT + 255) / 256;
    gat_init<<<initGrid, 256, 0, stream>>>(nh, m, denom, N);

    int rowTiles = (N + 15) / 16;
    gat_gemm_z<<<(rowTiles + 7) / 8, 256, 0, stream>>>(h, W_fc, z, N);

    gat_node_scores<<<(N + 7) / 8, 256, 0, stream>>>(z, w_attn, s_src, s_dst, N);

    gat_edge1<<<(E + 7) / 8, 256, 0, stream>>>(y, w_attn, s_src, s_dst, src, dst,
                                               e_buf, m, E);

    gat_edge2<<<(E + 255) / 256, 256, 0, stream>>>(e_buf, m, dst, denom, E);

    size_t p3threads = (size_t)E * DOUT;
    gat_edge3<<<(int)((p3threads + 255) / 256), 256, 0, stream>>>(e_buf, denom, z,
                                                                  src, dst, nh, E);
}